// SmallSMBlockPY_45492293599240
// MI455X (gfx1250) — compile-verified
//
#include <hip/hip_runtime.h>
#include <stdint.h>

// Fused per-pixel kernel-prediction stencil for MI455X (gfx1250).
// Bandwidth-bound (~68MB traffic, ~0.3 GFLOP) -> the win is data movement:
// all stencil operands are staged into LDS with GLOBAL_LOAD_ASYNC_TO_LDS_B128
// (ASYNCcnt), double-buffered across the 8 x-batches, gated by s_wait_asynccnt.
// Boundary handling is done in LDS (clamped-address stage + zero overwrite),
// so the compute path is branch-free and uniform.

typedef float v4f __attribute__((ext_vector_type(4)));

// Typed pointers for the async-to-LDS builtin (expects vector-of-int pointee).
typedef int vi4 __attribute__((vector_size(4 * sizeof(int))));
typedef __attribute__((address_space(1))) vi4 vi4_as1;
typedef __attribute__((address_space(3))) vi4 vi4_as3;

#define HH 1024
#define WW 1024
#define NB 8
#define RPITCH 1032          // cols -1..1024 at slot+4 => b128-aligned interior
#define RSZ (3 * RPITCH)     // one 3-row region

#if __has_builtin(__builtin_amdgcn_global_load_async_to_lds_b128)
#define USE_ASYNC 1
#if __has_builtin(__builtin_amdgcn_s_wait_asynccnt)
#define WAIT_ASYNC(n) __builtin_amdgcn_s_wait_asynccnt(n)
#else
#define WAIT_ASYNC(n) asm volatile("s_wait_asynccnt %0" ::"i"(n) : "memory")
#endif
#else
#define USE_ASYNC 0
#define WAIT_ASYNC(n) ((void)0)
#endif

// Stage rows h-1..h+1 (addresses clamped; OOB rows fixed up in LDS later).
__device__ __forceinline__ void stage3(const float* __restrict__ gbase, int h,
                                       float* region, int w0) {
#pragma unroll
  for (int r = 0; r < 3; ++r) {
    int ih = h - 1 + r;
    ih = ih < 0 ? 0 : (ih > HH - 1 ? HH - 1 : ih);
    const float* src = gbase + ih * WW + w0;   // 16B aligned
    float* dst = region + r * RPITCH + w0 + 4; // 16B aligned
#if USE_ASYNC
    __builtin_amdgcn_global_load_async_to_lds_b128(
        (vi4_as1*)src, (vi4_as3*)dst, /*offset=*/0, /*cpol=*/0);
#else
    *(v4f*)dst = *(const v4f*)src;
#endif
  }
}

// Overwrite out-of-image rows with zeros (uniform branch; after WAIT, before barrier).
__device__ __forceinline__ void zero_oob(float* region, int h, int w0) {
  const v4f z = {0.f, 0.f, 0.f, 0.f};
  if (h == 0)      *(v4f*)(region + 0 * RPITCH + w0 + 4) = z;
  if (h == HH - 1) *(v4f*)(region + 2 * RPITCH + w0 + 4) = z;
}

__global__ __launch_bounds__(256) void kpn_fused(
    const float* __restrict__ img,
    const float* __restrict__ x,
    const float* __restrict__ kw,   // (9,1,3,3)
    const float* __restrict__ kb,   // (9,)
    float* __restrict__ y)          // (8,1,H,W)
{
  __shared__ float lds[3 * RSZ];    // [image region | x buf0 | x buf1]
  const int tid = threadIdx.x;      // 0..255
  const int h   = blockIdx.x;       // one output row per block
  const int w0  = tid << 2;         // 4 pixels per thread

  float* rimg  = lds;
  float* rbuf0 = lds + RSZ;
  float* rbuf1 = lds + 2 * RSZ;

  // Halo columns (cols -1 and 1024) are outside the image for every row:
  // zero them once; async stages never touch those slots.
  if (tid < 3) {
#pragma unroll
    for (int g = 0; g < 3; ++g) {
      lds[g * RSZ + tid * RPITCH + 3]    = 0.f;
      lds[g * RSZ + tid * RPITCH + 1028] = 0.f;
    }
  }

  // Prologue: stage image rows, then x batch 0.
  stage3(img, h, rimg, w0);         // 3 async
  stage3(x, h, rbuf0, w0);          // 3 async
  WAIT_ASYNC(3);                    // in-order => image rows have landed
  zero_oob(rimg, h, w0);
  __syncthreads();

  // ---- Per-pixel predicted kernels from LDS image taps ----
  float t0[7], t1[7], t2[7];
#pragma unroll
  for (int c = 0; c < 7; ++c) {
    t0[c] = rimg[0 * RPITCH + w0 + 3 + c];
    t1[c] = rimg[1 * RPITCH + w0 + 3 + c];
    t2[c] = rimg[2 * RPITCH + w0 + 3 + c];
  }
  float K0[9], K1[9], K2[9], K3[9];
#pragma unroll
  for (int c = 0; c < 9; ++c) {
    float a0 = kb[c], a1 = kb[c], a2 = kb[c], a3 = kb[c];
    const float* wc = &kw[c * 9];   // wave-uniform -> scalar loads
#pragma unroll
    for (int d = 0; d < 3; ++d) {
      const float wv0 = wc[0 + d], wv1 = wc[3 + d], wv2 = wc[6 + d];
      a0 += wv0 * t0[0 + d] + wv1 * t1[0 + d] + wv2 * t2[0 + d];
      a1 += wv0 * t0[1 + d] + wv1 * t1[1 + d] + wv2 * t2[1 + d];
      a2 += wv0 * t0[2 + d] + wv1 * t1[2 + d] + wv2 * t2[2 + d];
      a3 += wv0 * t0[3 + d] + wv1 * t1[3 + d] + wv2 * t2[3 + d];
    }
    K0[c] = a0; K1[c] = a1; K2[c] = a2; K3[c] = a3;
  }

  // ---- Double-buffered async pipeline over the 8 x-batches ----
#pragma unroll
  for (int b = 0; b < NB; ++b) {
    float* cb = (b & 1) ? rbuf1 : rbuf0;
    float* nb = (b & 1) ? rbuf0 : rbuf1;
    if (b + 1 < NB) {
      stage3(x + (size_t)(b + 1) * (HH * WW), h, nb, w0);  // 3 async ahead
      WAIT_ASYNC(3);   // retire batch b's 3 rows (next batch still in flight)
    } else {
      WAIT_ASYNC(0);   // last batch: drain
    }
    zero_oob(cb, h, w0);
    __syncthreads();   // batch b fully visible to all waves

    float a0 = 0.f, a1 = 0.f, a2 = 0.f, a3 = 0.f;
#pragma unroll
    for (int r = 0; r < 3; ++r) {
      float q[6];
#pragma unroll
      for (int c = 0; c < 6; ++c) q[c] = cb[r * RPITCH + w0 + 3 + c];
#pragma unroll
      for (int d = 0; d < 3; ++d) {
        a0 += K0[r * 3 + d] * q[0 + d];
        a1 += K1[r * 3 + d] * q[1 + d];
        a2 += K2[r * 3 + d] * q[2 + d];
        a3 += K3[r * 3 + d] * q[3 + d];
      }
    }
    v4f out; out.x = a0; out.y = a1; out.z = a2; out.w = a3;
    *(v4f*)(y + (size_t)b * (HH * WW) + (size_t)h * WW + w0) = out;  // b128

    if (b + 1 < NB) __syncthreads();  // protect cb before it is re-staged
  }
}

extern "C" void kernel_launch(void* const* d_in, const int* in_sizes, int n_in,
                              void* d_out, int out_size, void* d_ws, size_t ws_size,
                              hipStream_t stream) {
  const float* img = (const float*)d_in[0];   // (1,1024,1024)
  const float* x   = (const float*)d_in[1];   // (8,1,1024,1024)
  const float* kw  = (const float*)d_in[2];   // (9,1,3,3)
  const float* kb  = (const float*)d_in[3];   // (9,)
  float* y = (float*)d_out;                   // (8,1,1024,1024)
  (void)in_sizes; (void)n_in; (void)out_size; (void)d_ws; (void)ws_size;

  kpn_fused<<<dim3(HH), dim3(256), 0, stream>>>(img, x, kw, kb, y);
}